// GCN_62534723830420
// MI455X (gfx1250) — compile-verified
//
#include <hip/hip_runtime.h>
#include <hip/hip_bf16.h>

typedef __attribute__((ext_vector_type(2))) float v2f;
typedef __attribute__((ext_vector_type(8))) float v8f;

// ---------------------------------------------------------------------------
// Dense GEMM + bias: out[N,F] = X[N,K] @ W[K,F] + b[F], via V_WMMA_F32_16X16X4_F32.
// blockDim.x = 32 * (F/16). Wave w computes the 16x16 tile at (blockIdx.x*16, w*16).
// N must be a multiple of 16 (100000 = 6250*16) so EXEC is all ones at every WMMA.
// ---------------------------------------------------------------------------
__global__ __launch_bounds__(256) void gcn_gemm_bias_wmma(
    const float* __restrict__ X, const float* __restrict__ W,
    const float* __restrict__ bias, float* __restrict__ out,
    int K, int F)
{
    const int wave = threadIdx.x >> 5;
    const int lane = threadIdx.x & 31;
    const int r0   = blockIdx.x * 16;
    const int c0   = wave * 16;

    const int m  = lane & 15;   // A row / B-D column within tile
    const int hi = lane >> 4;   // lane half selects K pair {0,1} vs {2,3}

    v8f acc = {0.f, 0.f, 0.f, 0.f, 0.f, 0.f, 0.f, 0.f};

    // A: lane holds X[r0+m, k0+2*hi] and X[r0+m, k0+2*hi+1]  (contiguous pair)
    const float* arow = X + (size_t)(r0 + m) * K + 2 * hi;
    // B: lane holds W[k0+2*hi, c0+m] and W[k0+2*hi+1, c0+m]
    const float* bcol = W + (size_t)(2 * hi) * F + c0 + m;

    for (int k0 = 0; k0 < K; k0 += 4) {
        v2f a, b;
        a.x = arow[k0];
        a.y = arow[k0 + 1];
        b.x = bcol[(size_t)k0 * F];
        b.y = bcol[(size_t)(k0 + 1) * F];
        // 8 args: (neg_a, A, neg_b, B, c_mod, C, reuse_a, reuse_b)
        acc = __builtin_amdgcn_wmma_f32_16x16x4_f32(
            false, a, false, b, (short)0, acc, false, false);
    }

    const float bv = bias[c0 + m];
    // D layout: lanes 0-15: VGPR i -> row i ; lanes 16-31: VGPR i -> row 8+i
    #pragma unroll
    for (int i = 0; i < 8; ++i) {
        int row = r0 + hi * 8 + i;
        out[(size_t)row * F + c0 + m] = acc[i] + bv;
    }
}

// ---------------------------------------------------------------------------
// SpMM scatter: out[row[e], :] += val[e] * H[col[e], :].
// One wave per edge; each lane covers F/32 contiguous features.
// H (<= 51.2 MB) is L2-resident on MI455X (192 MB L2), so gathers + f32
// atomics are absorbed by L2, never HBM.
// ---------------------------------------------------------------------------
__global__ __launch_bounds__(256) void gcn_spmm_atomic(
    const long long* __restrict__ rows, const long long* __restrict__ cols,
    const float* __restrict__ vals, const float* __restrict__ H,
    float* __restrict__ out, int E, int F)
{
    const int gwave = (int)((blockIdx.x * (unsigned)blockDim.x + threadIdx.x) >> 5);
    const int lane  = threadIdx.x & 31;
    if (gwave >= E) return;

    const long long r = rows[gwave];
    const long long c = cols[gwave];
    const float     v = vals[gwave];

    if (F == 128) {
        const float4* src = (const float4*)(H + (size_t)c * 128) + lane;
        float*        dst = out + (size_t)r * 128 + lane * 4;
        float4 x = *src;
        atomicAdd(dst + 0, v * x.x);
        atomicAdd(dst + 1, v * x.y);
        atomicAdd(dst + 2, v * x.z);
        atomicAdd(dst + 3, v * x.w);
    } else { // F == 64
        const float2* src = (const float2*)(H + (size_t)c * 64) + lane;
        float*        dst = out + (size_t)r * 64 + lane * 2;
        float2 x = *src;
        atomicAdd(dst + 0, v * x.x);
        atomicAdd(dst + 1, v * x.y);
    }
}

// ---------------------------------------------------------------------------
// Elementwise ReLU (hidden layers only).
// ---------------------------------------------------------------------------
__global__ __launch_bounds__(256) void gcn_relu(float* __restrict__ h, size_t n)
{
    size_t i = (size_t)blockIdx.x * blockDim.x + threadIdx.x;
    size_t stride = (size_t)gridDim.x * blockDim.x;
    for (; i < n; i += stride) {
        float x = h[i];
        h[i] = x > 0.f ? x : 0.f;
    }
}

extern "C" void kernel_launch(void* const* d_in, const int* in_sizes, int n_in,
                              void* d_out, int out_size, void* d_ws, size_t ws_size,
                              hipStream_t stream)
{
    const int IN_F = 128, HID_F = 128, OUT_F = 64;

    const float*     x    = (const float*)d_in[0];
    const long long* arow = (const long long*)d_in[1];   // int64 in reference
    const long long* acol = (const long long*)d_in[2];
    const float*     aval = (const float*)d_in[3];
    const float*     W0   = (const float*)d_in[4];
    const float*     b0   = (const float*)d_in[5];
    const float*     W1   = (const float*)d_in[6];
    const float*     b1   = (const float*)d_in[7];
    const float*     W2   = (const float*)d_in[8];
    const float*     b2   = (const float*)d_in[9];

    const int N = in_sizes[0] / IN_F;      // 100000 (multiple of 16)
    const int E = in_sizes[1];             // 1600000

    float* bufG = (float*)d_ws;                       // GEMM output   [N,128]
    float* bufS = bufG + (size_t)N * HID_F;           // SpMM output   [N,128]
    float* outp = (float*)d_out;                      // final         [N,64]

    const int rowTiles   = N / 16;                    // 6250
    const int spmmBlocks = (E * 32 + 255) / 256;
    const int reluBlocks = 2048;

    // ---- Layer 0: h = relu(spmm(x @ W0 + b0)) ----
    gcn_gemm_bias_wmma<<<rowTiles, 32 * (HID_F / 16), 0, stream>>>(x, W0, b0, bufG, IN_F, HID_F);
    hipMemsetAsync(bufS, 0, (size_t)N * HID_F * sizeof(float), stream);
    gcn_spmm_atomic<<<spmmBlocks, 256, 0, stream>>>(arow, acol, aval, bufG, bufS, E, HID_F);
    gcn_relu<<<reluBlocks, 256, 0, stream>>>(bufS, (size_t)N * HID_F);

    // ---- Layer 1: h = relu(spmm(h @ W1 + b1)) ----
    gcn_gemm_bias_wmma<<<rowTiles, 32 * (HID_F / 16), 0, stream>>>(bufS, W1, b1, bufG, HID_F, HID_F);
    hipMemsetAsync(bufS, 0, (size_t)N * HID_F * sizeof(float), stream);
    gcn_spmm_atomic<<<spmmBlocks, 256, 0, stream>>>(arow, acol, aval, bufG, bufS, E, HID_F);
    gcn_relu<<<reluBlocks, 256, 0, stream>>>(bufS, (size_t)N * HID_F);

    // ---- Layer 2: out = spmm(h @ W2 + b2) ----
    gcn_gemm_bias_wmma<<<rowTiles, 32 * (OUT_F / 16), 0, stream>>>(bufS, W2, b2, bufG, HID_F, OUT_F);
    hipMemsetAsync(outp, 0, (size_t)N * OUT_F * sizeof(float), stream);
    gcn_spmm_atomic<<<spmmBlocks, 256, 0, stream>>>(arow, acol, aval, bufG, outp, E, OUT_F);
}